// MultiBoxLoss_1391569404423
// MI455X (gfx1250) — compile-verified
//
#include <hip/hip_runtime.h>
#include <hip/hip_bf16.h>
#include <stdint.h>

// Problem constants (SSD300 / COCO per reference)
#define BATCH 64
#define NANCH 8732
#define NCLS  81
#define TILE_ANCHORS 16
#define TW (TILE_ANCHORS * NCLS)         // 1296 floats per tile
#define WAVES_PER_BLOCK 4                // 128-thread blocks (4 wave32)
#define TOTAL_ANCHORS (BATCH * NANCH)    // 558848 = 16 * 34928
#define TOTAL_TILES (TOTAL_ANCHORS / TILE_ANCHORS)      // 34928
#define GRID_CONF (TOTAL_TILES / WAVES_PER_BLOCK)       // 8732 blocks, exact

typedef float v2f __attribute__((ext_vector_type(2)));
typedef float v8f __attribute__((ext_vector_type(8)));
typedef unsigned int v4u __attribute__((ext_vector_type(4)));
typedef int v4i __attribute__((ext_vector_type(4)));
typedef int v8i __attribute__((ext_vector_type(8)));

#if defined(__has_builtin)
#if __has_builtin(__builtin_amdgcn_tensor_load_to_lds) && \
    __has_builtin(__builtin_amdgcn_s_wait_tensorcnt)
#define HAVE_TDM 1
#endif
#endif
#ifndef HAVE_TDM
#define HAVE_TDM 0
#endif

// ---------------------------------------------------------------------------
// Workspace layout (d_ws):
//   [0]    int   num_pos
//   [16]   float thr[BATCH]
//   [512]  float cl_neg[TOTAL_ANCHORS]   (neg_mask * conf_loss, 2.24 MB)
// ---------------------------------------------------------------------------

__global__ void init_kernel(float* __restrict__ out, int* __restrict__ num_pos) {
    int t = threadIdx.x;
    if (t < 2 * BATCH) out[t] = 0.0f;    // conf_total[64] + loc_total[64]
    if (t == 0) *num_pos = 0;
}

#if HAVE_TDM
// TDM D# for a 2D 16x81 f32 tile: contiguous rows in memory (stride 81) and
// contiguous destination in LDS. Packed per CDNA5 ISA Tensor-DMA descriptor:
//   group0: count=1 | lds_addr | global_addr[56:0] | type=2
//   group1: data_size=4B, tensor_dim0=81, tensor_dim1=16, tile_dim0=81,
//           tile_dim1=16, tensor_dim0_stride=81, no pad / iterate / multicast
__device__ __forceinline__ void tdm_load_tile_16x81(unsigned long long gaddr,
                                                    unsigned lds_off_bytes) {
    v4u g0;
    g0.x = 1u;                                            // count=1, is_restore=0
    g0.y = lds_off_bytes;                                 // lds_addr (bytes)
    g0.z = (unsigned)(gaddr & 0xFFFFFFFFu);               // global_addr[31:0]
    g0.w = (unsigned)((gaddr >> 32) & 0x01FFFFFFu)        // global_addr[56:32]
         | (2u << 30);                                    // type=2 ("image")
    // group1 assembled as 4x u64 quads (bit positions per ISA 8.4)
    unsigned long long q0 = (2ull << 16)                  // data_size = 4 bytes
                          | ((unsigned long long)NCLS << 48);          // tensor_dim0[15:0] @ bits 79:48
    unsigned long long q1 = ((unsigned long long)TILE_ANCHORS << 16)   // tensor_dim1 @ bits 111:80
                          | ((unsigned long long)NCLS << 48);          // tile_dim0   @ bits 127:112
    unsigned long long q2 = ((unsigned long long)TILE_ANCHORS)         // tile_dim1   @ bits 143:128
                          | ((unsigned long long)NCLS << 32);          // tensor_dim0_stride @ bits 207:160
    unsigned long long q3 = 0ull;                         // tensor_dim1_stride unused (2D)
    v8i g1;
    g1[0] = (int)(unsigned)(q0);  g1[1] = (int)(unsigned)(q0 >> 32);
    g1[2] = (int)(unsigned)(q1);  g1[3] = (int)(unsigned)(q1 >> 32);
    g1[4] = (int)(unsigned)(q2);  g1[5] = (int)(unsigned)(q2 >> 32);
    g1[6] = (int)(unsigned)(q3);  g1[7] = (int)(unsigned)(q3 >> 32);
    v4i z4 = {};
#if __clang_major__ >= 23
    v8i z8 = {};
    __builtin_amdgcn_tensor_load_to_lds(g0, g1, z4, z4, z8, 0);
#else
    __builtin_amdgcn_tensor_load_to_lds(g0, g1, z4, z4, 0);
#endif
}
#endif

// ---------------------------------------------------------------------------
// Kernel 2: per-anchor conf loss.  One wave = one 16-anchor tile.
//   tiles DMA'd to LDS via Tensor Data Mover (TENSORcnt-tracked)
//   lse  = logsumexp(pred_conf[anchor, 0:81])      (VALU, 2 lanes/anchor)
//   dot  = <pred_conf, gt_conf>  via V_WMMA_F32_16X16X4_F32 (D[t][t] for t<8, D[t][t-8] for t>=8)
//   cl   = lse - dot;  cl_neg = (1-pos)*cl  -> ws
// ---------------------------------------------------------------------------
__global__ __launch_bounds__(128) void conf_loss_kernel(
    const float* __restrict__ pred_conf, const float* __restrict__ gt_conf,
    float* __restrict__ cl_neg, float* __restrict__ conf_out,
    int* __restrict__ num_pos)
{
    __shared__ float lds[WAVES_PER_BLOCK * 2 * TW];   // 41.5 KB
    const int lane = threadIdx.x & 31;
    const int wave = threadIdx.x >> 5;
    const int tile = blockIdx.x * WAVES_PER_BLOCK + wave;
    const int anchor0 = tile * TILE_ANCHORS;

    float* sp = lds + wave * TW;
    float* sg = lds + WAVES_PER_BLOCK * TW + wave * TW;

    // Speculative prefetch of the next block's stream (global_prefetch_b8)
    __builtin_prefetch(pred_conf + (size_t)(anchor0 + WAVES_PER_BLOCK * TILE_ANCHORS) * NCLS, 0, 1);
    __builtin_prefetch(gt_conf  + (size_t)(anchor0 + WAVES_PER_BLOCK * TILE_ANCHORS) * NCLS, 0, 1);

#if HAVE_TDM
    // DMA both tiles straight into LDS (no VGPR round-trip), then fence.
    tdm_load_tile_16x81((unsigned long long)(uintptr_t)(pred_conf + (size_t)anchor0 * NCLS),
                        (unsigned)((const char*)sp - (const char*)lds));
    tdm_load_tile_16x81((unsigned long long)(uintptr_t)(gt_conf + (size_t)anchor0 * NCLS),
                        (unsigned)((const char*)sg - (const char*)lds));
    __builtin_amdgcn_s_wait_tensorcnt(0);
#else
    // Fallback: flat coalesced copy (tile is contiguous, stride == NCLS)
    {
        const float* pg = pred_conf + (size_t)anchor0 * NCLS;
        const float* gg = gt_conf  + (size_t)anchor0 * NCLS;
        for (int i = lane; i < TW; i += 32) { sp[i] = pg[i]; sg[i] = gg[i]; }
    }
#endif
    __syncthreads();

    // positive mask: gt background prob == 0  (read before sg gets reused)
    float gt0 = (lane < TILE_ANCHORS) ? sg[lane * NCLS] : 1.0f;
    bool pos = (gt0 == 0.0f);

    // logsumexp over 81 classes: 2 lanes per anchor
    const int r2 = lane >> 1, par = lane & 1;
    const float* lrow = sp + r2 * NCLS;
    float mx = -3.402823e38f;
    for (int c = par; c < NCLS; c += 2) mx = fmaxf(mx, lrow[c]);
    mx = fmaxf(mx, __shfl_xor(mx, 1, 32));
    float sum = 0.0f;
    for (int c = par; c < NCLS; c += 2) sum += __expf(lrow[c] - mx);
    sum += __shfl_xor(sum, 1, 32);
    float lse = mx + __logf(sum);

    // WMMA one-hot dot: D[i][j] = sum_k pred[a0+i][k] * gt[a0+j][k]
    // A 16x4 f32 layout: lanes<16 hold K=k0,k0+1; lanes>=16 hold K=k0+2,k0+3
    v8f acc = {};
    const int half = lane >> 4, mr = lane & 15;
    const float* arow = sp + mr * NCLS;
    const float* brow = sg + mr * NCLS;     // B[k][j] = gt[anchor j][k]
    for (int k0 = 0; k0 < 80; k0 += 4) {
        v2f a, b;
        a.x = arow[k0 + (half ? 2 : 0)];
        a.y = arow[k0 + (half ? 3 : 1)];
        b.x = brow[k0 + (half ? 2 : 0)];
        b.y = brow[k0 + (half ? 3 : 1)];
        acc = __builtin_amdgcn_wmma_f32_16x16x4_f32(
            false, a, false, b, (short)0, acc, false, false);
    }
    {   // tail K-step: only K=80 is real; K=81..83 are zeros
        v2f a, b;
        a.x = half ? 0.0f : arow[80];
        a.y = 0.0f;
        b.x = half ? 0.0f : brow[80];
        b.y = 0.0f;
        acc = __builtin_amdgcn_wmma_f32_16x16x4_f32(
            false, a, false, b, (short)0, acc, false, false);
    }
    __syncthreads();

    // Stash D and lse into (now-dead) gt tile region; pull D[t][t] (t<8) / D[t][t-8] (t>=8).
    #pragma unroll
    for (int r = 0; r < 8; ++r) sg[r * 32 + lane] = acc[r];
    if (par == 0) sg[256 + r2] = lse;
    __syncthreads();

    if (lane < TILE_ANCHORS) {
        const int t = lane;
        // C/D layout: VGPR r: lanes 0-15 -> (M=r, N=lane); lanes 16-31 -> (M=r+8)
        float diag = (t < 8) ? sg[t * 32 + t] : sg[(t - 8) * 32 + (t + 8)];
        float cl = sg[256 + t] - diag;           // lse[t] - <pred[t], gt[t>=8 ? t-8 : t]>
        int idx = anchor0 + t;
        int b = idx / NANCH;
        cl_neg[idx] = pos ? 0.0f : cl;
        if (pos) {
            atomicAdd(&conf_out[b], cl);         // positive_conf_loss part
            atomicAdd(num_pos, 1);
        }
    }
}

// ---------------------------------------------------------------------------
// Kernel 3: per-row stable rank-k select (k = int(3*num_pos)) of the
// argsort index, faithful to jnp.argsort stable ties: key = (value_bits, i).
// MSB-first 8x8-bit radix select over an LDS-resident row.
// (cl_neg >= 0 so IEEE bits are order-preserving.)
// ---------------------------------------------------------------------------
__global__ __launch_bounds__(256) void select_kernel(
    const float* __restrict__ cl_neg, const int* __restrict__ num_pos,
    float* __restrict__ thr)
{
    __shared__ float row[NANCH];                 // 34928 B
    __shared__ unsigned hist[256];
    __shared__ unsigned long long sh_prefix;
    __shared__ int sh_k;
    const int b = blockIdx.x;
    const int tid = threadIdx.x;

    for (int i = tid; i < NANCH; i += 256)
        row[i] = cl_neg[(size_t)b * NANCH + i];
    if (tid == 0) {
        sh_prefix = 0ull;
        sh_k = (int)(3.0f * (float)(*num_pos));  // num_neg (0-based rank)
    }
    __syncthreads();

    for (int shift = 56; shift >= 0; shift -= 8) {
        const unsigned long long pref = sh_prefix;
        const int kcur = sh_k;
        hist[tid] = 0u;
        __syncthreads();
        for (int i = tid; i < NANCH; i += 256) {
            unsigned vb = __float_as_uint(row[i]);
            unsigned long long key = ((unsigned long long)vb << 32) | (unsigned)i;
            bool match = (shift == 56) || ((key >> (shift + 8)) == pref);
            if (match) atomicAdd(&hist[(unsigned)((key >> shift) & 0xFF)], 1u);
        }
        __syncthreads();
        if (tid == 0) {
            int kk = kcur;
            for (int d = 0; d < 256; ++d) {
                int c = (int)hist[d];
                if (kk < c) { sh_prefix = (pref << 8) | (unsigned)d; sh_k = kk; break; }
                kk -= c;
            }
        }
        __syncthreads();
    }
    if (tid == 0)
        thr[b] = (float)(unsigned)(sh_prefix & 0xFFFFFFFFull); // order[b, num_neg]
}

// ---------------------------------------------------------------------------
// Kernel 4: hard-negative sum: sum of cl_neg where cl_neg > thr[b].
// Block covers 256 consecutive anchors -> at most 2 batch rows; LDS reduce.
// ---------------------------------------------------------------------------
__global__ __launch_bounds__(256) void neg_sum_kernel(
    const float* __restrict__ cl_neg, const float* __restrict__ thr,
    float* __restrict__ conf_out)
{
    __shared__ float acc2[2];
    const int i = blockIdx.x * 256 + threadIdx.x;
    const int b_first = (blockIdx.x * 256) / NANCH;
    if (threadIdx.x < 2) acc2[threadIdx.x] = 0.0f;
    __syncthreads();
    const int b = i / NANCH;
    float v = cl_neg[i];
    if (v > thr[b]) atomicAdd(&acc2[b - b_first], v);
    __syncthreads();
    if (threadIdx.x < 2) {
        int bb = b_first + threadIdx.x;
        if (bb < BATCH && acc2[threadIdx.x] != 0.0f)
            atomicAdd(&conf_out[bb], acc2[threadIdx.x]);
    }
}

// ---------------------------------------------------------------------------
// Kernel 5: loc loss: (|d| - 0.5) where |d| > 1, else 0 (reference's masked
// smooth-L1 quirk). One thread per loc element; LDS per-row reduce.
// ---------------------------------------------------------------------------
__global__ __launch_bounds__(256) void loc_loss_kernel(
    const float* __restrict__ pred_loc, const float* __restrict__ gt_loc,
    float* __restrict__ loc_out)
{
    __shared__ float acc2[2];
    const int i = blockIdx.x * 256 + threadIdx.x;
    const int row_elems = NANCH * 4;
    const int b_first = (blockIdx.x * 256) / row_elems;
    if (threadIdx.x < 2) acc2[threadIdx.x] = 0.0f;
    __syncthreads();
    const int b = i / row_elems;
    float d = pred_loc[i] - gt_loc[i];
    float a = fabsf(d);
    if (a > 1.0f) atomicAdd(&acc2[b - b_first], a - 0.5f);
    __syncthreads();
    if (threadIdx.x < 2) {
        int bb = b_first + threadIdx.x;
        if (bb < BATCH && acc2[threadIdx.x] != 0.0f)
            atomicAdd(&loc_out[bb], acc2[threadIdx.x]);
    }
}

// ---------------------------------------------------------------------------
extern "C" void kernel_launch(void* const* d_in, const int* in_sizes, int n_in,
                              void* d_out, int out_size, void* d_ws, size_t ws_size,
                              hipStream_t stream) {
    const float* pred_conf = (const float*)d_in[0];   // [B,N,C]
    const float* pred_loc  = (const float*)d_in[1];   // [B,N,4]
    const float* gt_conf   = (const float*)d_in[2];   // [B,N,C]
    const float* gt_loc    = (const float*)d_in[3];   // [B,N,4]
    float* out = (float*)d_out;                       // [0:64]=conf, [64:128]=loc

    char* ws = (char*)d_ws;
    int*   num_pos = (int*)ws;
    float* thr     = (float*)(ws + 16);
    float* cl_neg  = (float*)(ws + 512);

    init_kernel<<<1, 128, 0, stream>>>(out, num_pos);
    conf_loss_kernel<<<GRID_CONF, 128, 0, stream>>>(pred_conf, gt_conf,
                                                    cl_neg, out, num_pos);
    select_kernel<<<BATCH, 256, 0, stream>>>(cl_neg, num_pos, thr);
    neg_sum_kernel<<<TOTAL_ANCHORS / 256, 256, 0, stream>>>(cl_neg, thr, out);
    loc_loss_kernel<<<(TOTAL_ANCHORS * 4) / 256, 256, 0, stream>>>(pred_loc, gt_loc,
                                                                   out + BATCH);
}